// SimplifiedBioelectricCore_59382217834966
// MI455X (gfx1250) — compile-verified
//
#include <hip/hip_runtime.h>
#include <hip/hip_bf16.h>
#include <math.h>

// Problem constants (reference: H = W = 4096, fp32).
#define GH 4096
#define GW 4096
#define TH 4                 // output rows per block
#define LROWS (TH + 2)       // staged rows incl. top/bottom halo
#define BLK 256              // 8 waves (wave32)
#define W4 (GW / 4)          // float4s per row = 1024

__device__ __forceinline__ float fast_tanh(float x) {
#if __has_builtin(__builtin_amdgcn_tanhf)
    return __builtin_amdgcn_tanhf(x);   // gfx1250 v_tanh_f32
#else
    return tanhf(x);
#endif
}

// Elementwise bioelectric update for one cell.
__device__ __forceinline__ void cell_update(float v, float lap, float na, float k, float ca,
                                            float& ov, float& ona, float& ok, float& oca) {
    float upd = 0.5f * lap + 0.1f * na - 0.05f * k + 0.08f * ca;
    float nv  = fast_tanh(v + 0.1f * upd);
    ov  = nv;
    ona = na * 0.95f + 0.05f * fmaxf(nv, 0.0f);
    ok  = k  * 0.95f + 0.05f * fmaxf(-nv, 0.0f);
    oca = ca * 0.95f + 0.05f * fabsf(nv);
}

__global__ __launch_bounds__(BLK) void bioelectric_stencil_kernel(
    const float* __restrict__ V,
    const float* __restrict__ Na,
    const float* __restrict__ K,
    const float* __restrict__ Ca,
    float* __restrict__ out)
{
    extern __shared__ float lds[];            // LROWS * GW floats = 96 KB
    const int tid = threadIdx.x;
    const int r0  = blockIdx.x * TH;          // first output row of this block
    const size_t N = (size_t)GH * GW;

    // ---- Stage LROWS full voltage rows into LDS via async copy engine ----
    // LDS destination address comes from the vdst VGPR; low 32 bits of a generic LDS pointer
    // are the workgroup-relative LDS byte offset on gfx1250.
#pragma unroll
    for (int t = 0; t < LROWS; ++t) {
        int gr = r0 - 1 + t;                  // reflect (mirror excl. edge)
        gr = (gr < 0) ? 1 : gr;
        gr = (gr >= GH) ? (GH - 2) : gr;
        const float* grow = V + (size_t)gr * GW;
        float*       lrow = lds + t * GW;
#pragma unroll
        for (int m = 0; m < W4 / BLK; ++m) {  // 4 float4s per thread per row
            int c = (tid + m * BLK) << 2;     // column (multiple of 4)
            unsigned lds_off = (unsigned)(uintptr_t)(lrow + c);
            unsigned long long ga = (unsigned long long)(uintptr_t)(grow + c);
            asm volatile("global_load_async_to_lds_b128 %0, %1, off"
                         :: "v"(lds_off), "v"(ga)
                         : "memory");
        }
    }
    // Wait for this wave's async copies (ASYNCcnt), then make the tile
    // visible across all 8 waves of the workgroup.
    asm volatile("s_wait_asynccnt 0" ::: "memory");
    __syncthreads();

    const float4* Na4 = (const float4*)Na;
    const float4* K4  = (const float4*)K;
    const float4* Ca4 = (const float4*)Ca;
    float4* outV  = (float4*)(out);
    float4* outNa = (float4*)(out + N);
    float4* outK  = (float4*)(out + 2 * N);
    float4* outCa = (float4*)(out + 3 * N);

    // ---- Compute TH rows; all voltage neighbor reuse comes from LDS ----
#pragma unroll
    for (int t = 0; t < TH; ++t) {
        const float* Lu = lds + (t + 0) * GW;   // row above
        const float* Lc = lds + (t + 1) * GW;   // center row
        const float* Ld = lds + (t + 2) * GW;   // row below
        const size_t rowBase4 = (size_t)(r0 + t) * W4;
#pragma unroll
        for (int m = 0; m < W4 / BLK; ++m) {
            int c4 = tid + m * BLK;             // float4 index in row
            int c  = c4 << 2;                   // element column

            float4 ctr = *(const float4*)(Lc + c);
            float4 up  = *(const float4*)(Lu + c);
            float4 dn  = *(const float4*)(Ld + c);
            // Column reflect: col0's left = col1; col4095's right = col4094.
            float lx = (c4 == 0)      ? ctr.y : Lc[c - 1];
            float rw = (c4 == W4 - 1) ? ctr.z : Lc[c + 4];

            float4 lap;
            lap.x = up.x + dn.x + lx    + ctr.y - 4.0f * ctr.x;
            lap.y = up.y + dn.y + ctr.x + ctr.z - 4.0f * ctr.y;
            lap.z = up.z + dn.z + ctr.y + ctr.w - 4.0f * ctr.z;
            lap.w = up.w + dn.w + ctr.z + rw    - 4.0f * ctr.w;

            size_t gi = rowBase4 + c4;
            float4 na = Na4[gi];
            float4 kk = K4[gi];
            float4 ca = Ca4[gi];

            float4 ov, ona, ok, oca;
            cell_update(ctr.x, lap.x, na.x, kk.x, ca.x, ov.x, ona.x, ok.x, oca.x);
            cell_update(ctr.y, lap.y, na.y, kk.y, ca.y, ov.y, ona.y, ok.y, oca.y);
            cell_update(ctr.z, lap.z, na.z, kk.z, ca.z, ov.z, ona.z, ok.z, oca.z);
            cell_update(ctr.w, lap.w, na.w, kk.w, ca.w, ov.w, ona.w, ok.w, oca.w);

            outV[gi]  = ov;
            outNa[gi] = ona;
            outK[gi]  = ok;
            outCa[gi] = oca;
        }
    }
}

extern "C" void kernel_launch(void* const* d_in, const int* in_sizes, int n_in,
                              void* d_out, int out_size, void* d_ws, size_t ws_size,
                              hipStream_t stream) {
    (void)in_sizes; (void)n_in; (void)out_size; (void)d_ws; (void)ws_size;
    const float* V  = (const float*)d_in[0];
    const float* Na = (const float*)d_in[1];
    const float* K  = (const float*)d_in[2];
    const float* Ca = (const float*)d_in[3];
    float* out = (float*)d_out;

    dim3 grid(GH / TH);                         // 1024 blocks
    dim3 block(BLK);                            // 256 threads = 8 wave32
    size_t shmem = (size_t)LROWS * GW * sizeof(float);  // 96 KB dynamic LDS
    bioelectric_stencil_kernel<<<grid, block, shmem, stream>>>(V, Na, K, Ca, out);
}